// SAM_73847667687770
// MI455X (gfx1250) — compile-verified
//
#include <hip/hip_runtime.h>

// ---------------- problem constants (from reference) ----------------
#define NBATCH 16
#define CIN    256
#define Hdim   56
#define Wdim   56
#define HW     (Hdim * Wdim)      // 3136
#define MROWS  96                 // 16 (conv1) + 16 (conv2) + 64 (conv3)
#define RELC   16
#define RELP   18                 // rel + 2 position channels
#define OUTC   64
#define CW     8                  // out // share_planes
#define COLT   (HW / 16)          // 196 column tiles per image
#define LDSTR  257                // padded LDS row stride (dwords) -> conflict free

typedef __attribute__((ext_vector_type(2))) float v2f;
typedef __attribute__((ext_vector_type(8))) float v8f;

// =====================================================================
// Kernel 1: fused 1x1 convs as WMMA f32 GEMM
//   y[n, r, l] = sum_c W[r, c] * x[n, c, l]    (r in 0..95)
// Each block: 8 waves, one 16-column tile per wave, 48 of the 96 output
// rows (blockIdx.y selects the half). Weights staged in padded LDS.
// WMMA 16x16x4 f32 layouts (ISA 7.12.2):
//   A 16x4 : lanes 0-15 -> M=lane, v[0]=K0 v[1]=K1 ; lanes 16-31 -> K2/K3
//   B 4x16 : lanes 0-15 -> N=lane, v[0]=K0 v[1]=K1 ; lanes 16-31 -> K2/K3
//   C/D    : VGPR r, lanes 0-15 -> M=r, N=lane ; lanes 16-31 -> M=r+8
// =====================================================================
__global__ __launch_bounds__(256) void sam_gemm_wmma(
    const float* __restrict__ x,   // [N, 256, HW]
    const float* __restrict__ w1,  // [16, 256]
    const float* __restrict__ w2,  // [16, 256]
    const float* __restrict__ w3,  // [64, 256]
    float* __restrict__ y)         // [N, 96, HW]
{
    __shared__ float wlds[48 * LDSTR];   // 49,344 B

    const int tid   = threadIdx.x;
    const int mBase = blockIdx.y * 48;   // 0 or 48

    // stage 48 rows of the fused weight matrix into padded LDS
    for (int idx = tid; idx < 48 * CIN; idx += 256) {
        const int rr = idx >> 8;          // local row 0..47
        const int c  = idx & 255;
        const int r  = mBase + rr;        // global row 0..95
        float v;
        if (r < 16)      v = w1[r * CIN + c];
        else if (r < 32) v = w2[(r - 16) * CIN + c];
        else             v = w3[(r - 32) * CIN + c];
        wlds[rr * LDSTR + c] = v;
    }
    __syncthreads();

    const int wave = tid >> 5;
    const int lane = tid & 31;
    const int half = lane >> 4;           // 0: K pair {0,1}, 1: {2,3}
    const int nlan = lane & 15;           // M index for A, N index for B

    const int colTile = blockIdx.x * 8 + wave;     // 0..3135
    const int n       = colTile / COLT;
    const int col0    = (colTile % COLT) * 16;

    const float* bbase = x + (size_t)n * CIN * HW + col0 + nlan;

    v8f acc0 = {}, acc1 = {}, acc2 = {};

    for (int kb = 0; kb < CIN; kb += 4) {
        const int k0 = kb + 2 * half;
        v2f b;
        b.x = bbase[(size_t)k0 * HW];
        b.y = bbase[(size_t)(k0 + 1) * HW];
        const float* al = &wlds[nlan * LDSTR + k0];

        v2f a0, a1, a2;
        a0.x = al[0 * 16 * LDSTR]; a0.y = al[0 * 16 * LDSTR + 1];
        a1.x = al[1 * 16 * LDSTR]; a1.y = al[1 * 16 * LDSTR + 1];
        a2.x = al[2 * 16 * LDSTR]; a2.y = al[2 * 16 * LDSTR + 1];

        acc0 = __builtin_amdgcn_wmma_f32_16x16x4_f32(false, a0, false, b,
                                                     (short)0, acc0, false, false);
        acc1 = __builtin_amdgcn_wmma_f32_16x16x4_f32(false, a1, false, b,
                                                     (short)0, acc1, false, false);
        acc2 = __builtin_amdgcn_wmma_f32_16x16x4_f32(false, a2, false, b,
                                                     (short)0, acc2, false, false);
    }

    // store: row = mBase + mt*16 + r + 8*half, col = col0 + nlan
    float* yb = y + (size_t)n * MROWS * HW + col0 + nlan;
#pragma unroll
    for (int r = 0; r < 8; ++r)
        yb[(size_t)(mBase + 0 * 16 + r + 8 * half) * HW] = acc0[r];
#pragma unroll
    for (int r = 0; r < 8; ++r)
        yb[(size_t)(mBase + 1 * 16 + r + 8 * half) * HW] = acc1[r];
#pragma unroll
    for (int r = 0; r < 8; ++r)
        yb[(size_t)(mBase + 2 * 16 + r + 8 * half) * HW] = acc2[r];
}

// =====================================================================
// Kernel 2: fused subtraction / BN / MLP / softmax / aggregation
// One thread per (n, pixel).
// =====================================================================
__device__ __forceinline__ int refl(int i, int n) {
    return i < 0 ? -i : (i >= n ? 2 * n - 2 - i : i);
}

__global__ __launch_bounds__(256) void sam_attn(
    const float* __restrict__ y,      // [N, 96, HW]  (x1 | x2 | x3)
    const float* __restrict__ convp,  // [2,2]
    const float* __restrict__ cw1,    // [16,18]
    const float* __restrict__ cw2,    // [8,16]
    const float* __restrict__ g1, const float* __restrict__ b1,
    const float* __restrict__ m1, const float* __restrict__ v1,
    const float* __restrict__ g2, const float* __restrict__ b2,
    const float* __restrict__ m2, const float* __restrict__ v2,
    float* __restrict__ out)          // [N, 64, HW]
{
    __shared__ float s_cw1[RELC * RELP];
    __shared__ float s_cw2[CW * RELC];
    __shared__ float s_cp[4];

    const int tid = threadIdx.x;
    for (int i = tid; i < RELC * RELP; i += 256) s_cw1[i] = cw1[i];
    for (int i = tid; i < CW * RELC;  i += 256) s_cw2[i] = cw2[i];
    if (tid < 4) s_cp[tid] = convp[tid];
    __syncthreads();

    const int gid = blockIdx.x * 256 + tid;   // 0 .. N*HW-1
    const int n = gid / HW;
    const int l = gid % HW;
    const int h = l / Wdim;
    const int w = l % Wdim;

    const float sx = 2.0f / (float)(Wdim - 1);
    const float sy = 2.0f / (float)(Hdim - 1);
    const float lw = -1.0f + (float)w * sx;
    const float lh = -1.0f + (float)h * sy;
    const float p0 = s_cp[0] * lw + s_cp[1] * lh;
    const float p1 = s_cp[2] * lw + s_cp[3] * lh;

    const float s1v = g1[l] * rsqrtf(v1[l] + 1e-3f);
    const float t1v = b1[l] - m1[l] * s1v;
    const float s2v = g2[l] * rsqrtf(v2[l] + 1e-3f);
    const float t2v = b2[l] - m2[l] * s2v;

    const float* ybase = y + (size_t)n * MROWS * HW;
    const float* x1p = ybase;              // rows  0..15
    const float* x2p = ybase + 16 * HW;    // rows 16..31
    const float* x3p = ybase + 32 * HW;    // rows 32..95

    float x1v[RELC];
#pragma unroll
    for (int c = 0; c < RELC; ++c) x1v[c] = x1p[(size_t)c * HW + l];

    int nl[9];
#pragma unroll
    for (int ki = 0; ki < 3; ++ki)
#pragma unroll
        for (int kj = 0; kj < 3; ++kj) {
            const int hh = refl(h + ki - 1, Hdim);
            const int ww = refl(w + kj - 1, Wdim);
            nl[ki * 3 + kj] = hh * Wdim + ww;
        }

    float h3[9][CW];
#pragma unroll
    for (int j = 0; j < 9; ++j) {
        const int lj = nl[j];
        float h1[RELP];
#pragma unroll
        for (int c = 0; c < RELC; ++c) {
            const float d = x1v[c] - x2p[(size_t)c * HW + lj];
            const float t = d * s1v + t1v;
            h1[c] = t > 0.f ? t : 0.f;
        }
        {
            const int hj = lj / Wdim, wj = lj % Wdim;
            const float lwj = -1.0f + (float)wj * sx;
            const float lhj = -1.0f + (float)hj * sy;
            const float d0 = p0 - (s_cp[0] * lwj + s_cp[1] * lhj);
            const float d1 = p1 - (s_cp[2] * lwj + s_cp[3] * lhj);
            const float u0 = d0 * s1v + t1v;
            const float u1 = d1 * s1v + t1v;
            h1[16] = u0 > 0.f ? u0 : 0.f;
            h1[17] = u1 > 0.f ? u1 : 0.f;
        }
        float h2[RELC];
#pragma unroll
        for (int o = 0; o < RELC; ++o) {
            float s = 0.f;
#pragma unroll
            for (int c = 0; c < RELP; ++c) s = fmaf(s_cw1[o * RELP + c], h1[c], s);
            const float t = s * s2v + t2v;
            h2[o] = t > 0.f ? t : 0.f;
        }
#pragma unroll
        for (int q = 0; q < CW; ++q) {
            float s = 0.f;
#pragma unroll
            for (int c = 0; c < RELC; ++c) s = fmaf(s_cw2[q * RELC + c], h2[c], s);
            h3[j][q] = s;
        }
    }

    // softmax over the 9 window taps, per weight channel q
#pragma unroll
    for (int q = 0; q < CW; ++q) {
        float mx = h3[0][q];
#pragma unroll
        for (int j = 1; j < 9; ++j) mx = fmaxf(mx, h3[j][q]);
        float sum = 0.f;
#pragma unroll
        for (int j = 0; j < 9; ++j) {
            const float e = expf(h3[j][q] - mx);
            h3[j][q] = e;
            sum += e;
        }
        const float inv = 1.0f / sum;
#pragma unroll
        for (int j = 0; j < 9; ++j) h3[j][q] *= inv;
    }

    // grouped weighted aggregation: out channel c uses weight channel c % 8
    float* ob = out + (size_t)n * OUTC * HW + l;
#pragma unroll
    for (int g = 0; g < 8; ++g) {
        float acc[CW];
#pragma unroll
        for (int q = 0; q < CW; ++q) acc[q] = 0.f;
#pragma unroll
        for (int j = 0; j < 9; ++j) {
            const int lj = nl[j];
#pragma unroll
            for (int q = 0; q < CW; ++q)
                acc[q] = fmaf(h3[j][q], x3p[(size_t)(g * CW + q) * HW + lj], acc[q]);
        }
#pragma unroll
        for (int q = 0; q < CW; ++q)
            ob[(size_t)(g * CW + q) * HW] = acc[q];
    }
}

// =====================================================================
extern "C" void kernel_launch(void* const* d_in, const int* in_sizes, int n_in,
                              void* d_out, int out_size, void* d_ws, size_t ws_size,
                              hipStream_t stream) {
    (void)in_sizes; (void)n_in; (void)out_size; (void)ws_size;
    const float* x    = (const float*)d_in[0];
    const float* w1   = (const float*)d_in[1];
    const float* w2   = (const float*)d_in[2];
    const float* w3   = (const float*)d_in[3];
    const float* cp   = (const float*)d_in[4];
    const float* cw1  = (const float*)d_in[5];
    const float* cw2  = (const float*)d_in[6];
    const float* g1   = (const float*)d_in[7];
    const float* b1   = (const float*)d_in[8];
    const float* m1   = (const float*)d_in[9];
    const float* v1   = (const float*)d_in[10];
    const float* g2   = (const float*)d_in[11];
    const float* b2   = (const float*)d_in[12];
    const float* m2   = (const float*)d_in[13];
    const float* v2   = (const float*)d_in[14];

    float* y = (float*)d_ws;   // [N, 96, HW] = 19.3 MB scratch

    // GEMM: 3136 column tiles / 8 waves per block = 392 blocks, x2 row halves
    dim3 grid1(NBATCH * COLT / 8, 2, 1);
    sam_gemm_wmma<<<grid1, 256, 0, stream>>>(x, w1, w2, w3, y);

    // epilogue: one thread per pixel
    dim3 grid2(NBATCH * HW / 256, 1, 1);
    sam_attn<<<grid2, 256, 0, stream>>>(y, cp, cw1, cw2,
                                        g1, b1, m1, v1, g2, b2, m2, v2,
                                        (float*)d_out);
}